// MDNFullCov_88862873354692
// MI455X (gfx1250) — compile-verified
//
#include <hip/hip_runtime.h>
#include <hip/hip_bf16.h>
#include <math.h>

// ---------------- problem constants (match reference) ----------------
#define BB      8192
#define DD      32
#define ADIM    16
#define KK      5
#define HID     512
#define NTRIL   528          // 32*33/2
#define INDIM   (DD + ADIM)  // 48
#define OUT_MU  (KK * DD)    // 160
#define OUT_L   (KK * NTRIL) // 2640
#define LOG2PI  1.8378770664093453

// GEMM tiling
#define BK      16           // K-chunk staged per step
#define LDB     18           // padded per-column k-stride in LDS (floats)

typedef float v2f __attribute__((ext_vector_type(2)));
typedef float v8f __attribute__((ext_vector_type(8)));

// ---------------------------------------------------------------------
// Pack X = concat(s, a)  ->  [B, 48]
// ---------------------------------------------------------------------
__global__ void pack_x_kernel(const float* __restrict__ s,
                              const float* __restrict__ a,
                              float* __restrict__ X) {
    int idx = blockIdx.x * blockDim.x + threadIdx.x;
    if (idx >= BB * INDIM) return;
    int b = idx / INDIM;
    int c = idx - b * INDIM;
    X[idx] = (c < DD) ? s[b * DD + c] : a[b * ADIM + (c - DD)];
}

// ---------------------------------------------------------------------
// GEMM helpers (all buffer selection is compile-time: no indexed moves)
// ---------------------------------------------------------------------
__device__ __forceinline__ void load_a_frag(v2f (&ar)[4], const float* __restrict__ Ak, int kb) {
#pragma unroll
    for (int kk = 0; kk < 4; ++kk)
        ar[kk] = *(const v2f*)(Ak + kk * 4 + kb);
}

__device__ __forceinline__ void load_b_regs(float (&br)[8], const float* __restrict__ Wcol,
                                            int N, int chunk, int sk) {
    const float* __restrict__ Wk = Wcol + (size_t)chunk * BK * N;
#pragma unroll
    for (int j = 0; j < 8; ++j)
        br[j] = Wk[(size_t)(sk + j) * N];
}

__device__ __forceinline__ void store_b_lds(float* sBn, const float (&br)[8], int sn, int sk) {
#pragma unroll
    for (int j = 0; j < 8; ++j)
        sBn[sn * LDB + sk + j] = br[j];
}

template <int NV>
__device__ __forceinline__ void compute_chunk(v8f (&acc)[4], const v2f (&ar)[4],
                                              const float* sBc, int lane, int kb) {
#pragma unroll
    for (int kk = 0; kk < 4; ++kk) {
#pragma unroll
        for (int t = 0; t < NV; ++t) {
            v2f bf = *(const v2f*)(sBc + ((lane & 15) + t * 16) * LDB + kk * 4 + kb);
            acc[t] = __builtin_amdgcn_wmma_f32_16x16x4_f32(
                         false, ar[kk], false, bf, (short)0, acc[t], false, false);
        }
    }
}

// ---------------------------------------------------------------------
// WMMA fp32 GEMM body. Block = 4 waves = 64(M) x 64(N) output block.
// Wave w owns M-rows [mtile*16, +16); all 4 waves share the B chunk
// (BK x 64) staged transposed into LDS ( sB[col][k], stride LDB ).
// Double-buffered with the chunk loop manually unrolled 2x so both the
// A-fragment registers and the LDS half are selected at compile time.
// NV = number of valid 16-wide N tiles (compile-time): WMMA loop is
// guard-free and EXEC stays all-ones.
// ---------------------------------------------------------------------
template <int NV>
__device__ __forceinline__ void gemm_body(const float* __restrict__ A,
                                          const float* __restrict__ W,
                                          const float* __restrict__ bias,
                                          float* __restrict__ Out,
                                          int N, int K, int apply_silu,
                                          int ntile0, int mtile,
                                          int lane, int tid, float* sB) {
    const int mrow  = lane & 15;
    const int kb    = (lane >> 4) * 2;                // 0 or 2
    const int nbase = ntile0 * 16;
    const float* __restrict__ Arow = A + (size_t)(mtile * 16 + mrow) * K;

    // cooperative B staging: thread -> (column sn, k-slice sk..sk+7)
    const int sn = tid & 63;                          // 0..63
    const int sk = (tid >> 6) * 8;                    // 0 or 8
    int gcol = nbase + sn;
    if (NV < 4) gcol = min(gcol, N - 1);              // stay in-bounds of W
    const float* __restrict__ Wcol = W + gcol;        // (k, gcol) at Wcol[k*N]

    float* sB0 = sB;
    float* sB1 = sB + 64 * LDB;

    v8f acc[4] = {{}, {}, {}, {}};
    v2f aregA[4], aregB[4];
    float breg[8];

    const int nchunks = K / BK;

    // ---- preload chunk 0 into aregA + sB0 ----
    load_a_frag(aregA, Arow, kb);
    load_b_regs(breg, Wcol, N, 0, sk);
    store_b_lds(sB0, breg, sn, sk);
    __syncthreads();

    int c = 0;
    for (;;) {
        // ---- even chunk: compute from aregA / sB0, prefetch into B-side ----
        {
            const bool more = (c + 1 < nchunks);
            if (more) {
                load_a_frag(aregB, Arow + (c + 1) * BK, kb);
                load_b_regs(breg, Wcol, N, c + 1, sk);
            }
            compute_chunk<NV>(acc, aregA, sB0, lane, kb);
            __syncthreads();
            if (more) store_b_lds(sB1, breg, sn, sk);
            __syncthreads();
            if (++c == nchunks) break;
        }
        // ---- odd chunk: compute from aregB / sB1, prefetch into A-side ----
        {
            const bool more = (c + 1 < nchunks);
            if (more) {
                load_a_frag(aregA, Arow + (c + 1) * BK, kb);
                load_b_regs(breg, Wcol, N, c + 1, sk);
            }
            compute_chunk<NV>(acc, aregB, sB1, lane, kb);
            __syncthreads();
            if (more) store_b_lds(sB0, breg, sn, sk);
            __syncthreads();
            if (++c == nchunks) break;
        }
    }

    // ---- epilogue: bias + optional SiLU ----
    const int rowbase = mtile * 16 + ((lane >> 4) ? 8 : 0);
#pragma unroll
    for (int t = 0; t < NV; ++t) {
        int col = nbase + t * 16 + (lane & 15);
        float bc = bias[col];
#pragma unroll
        for (int r = 0; r < 8; ++r) {
            float v = acc[t][r] + bc;
            if (apply_silu) v = v / (1.0f + __expf(-v));
            Out[(size_t)(rowbase + r) * N + col] = v;
        }
    }
}

__global__ __launch_bounds__(128)
void gemm_wmma_f32_kernel(const float* __restrict__ A,
                          const float* __restrict__ W,
                          const float* __restrict__ bias,
                          float* __restrict__ Out,
                          int N, int K, int apply_silu) {
    __shared__ float sB[2 * 64 * LDB];
    const int wave = threadIdx.x >> 5;                // 4 waves / block
    const int lane = threadIdx.x & 31;
    const int tid  = threadIdx.x;
    const int ntile0 = blockIdx.x * 4;                // first 16-wide N tile
    const int mtile  = blockIdx.y * 4 + wave;         // this wave's M tile
    const int nv = min(4, N / 16 - ntile0);           // block-uniform

    switch (nv) {                                     // scalar branch only
    case 4: gemm_body<4>(A, W, bias, Out, N, K, apply_silu, ntile0, mtile, lane, tid, sB); break;
    case 3: gemm_body<3>(A, W, bias, Out, N, K, apply_silu, ntile0, mtile, lane, tid, sB); break;
    case 2: gemm_body<2>(A, W, bias, Out, N, K, apply_silu, ntile0, mtile, lane, tid, sB); break;
    default: gemm_body<1>(A, W, bias, Out, N, K, apply_silu, ntile0, mtile, lane, tid, sB); break;
    }
}

// ---------------------------------------------------------------------
// Tiny head: logits[B,5] = h3 @ Wpi + bpi (42 MFLOP, plain VALU)
// ---------------------------------------------------------------------
__global__ void logits_kernel(const float* __restrict__ h3,
                              const float* __restrict__ Wpi,
                              const float* __restrict__ bpi,
                              float* __restrict__ logits) {
    int idx = blockIdx.x * blockDim.x + threadIdx.x;
    if (idx >= BB * KK) return;
    int b = idx / KK;
    int kk = idx - b * KK;
    const float* __restrict__ hrow = h3 + (size_t)b * HID;
    float acc = bpi[kk];
    for (int j = 0; j < HID; ++j)
        acc += hrow[j] * Wpi[(size_t)j * KK + kk];
    logits[idx] = acc;
}

// ---------------------------------------------------------------------
// Per-(b,k) Gaussian log prob: fused stable_tril + fp64 forward
// substitution (matches reference: fp32 transforms, fp64 solve).
// ---------------------------------------------------------------------
__global__ void mdn_comp_kernel(const float* __restrict__ sp,
                                const float* __restrict__ mu,
                                const float* __restrict__ Lv,
                                float* __restrict__ comp) {
    int idx = blockIdx.x * blockDim.x + threadIdx.x;
    if (idx >= BB * KK) return;
    int b = idx / KK;
    const float* __restrict__ lrow  = Lv + (size_t)idx * NTRIL;
    const float* __restrict__ murow = mu + (size_t)idx * DD;
    const float* __restrict__ sprow = sp + (size_t)b * DD;

    double y[DD];
    double logdet = 0.0, quad = 0.0;
    int base = 0;
    for (int i = 0; i < DD; ++i) {
        float draw = lrow[base + i];
        float spv  = (draw > 20.0f) ? draw : log1pf(__expf(draw));
        float dpos = fminf(spv + 1e-3f, 50.0f);

        double acc = (double)sprow[i] - (double)murow[i];
        float off_scale = 0.5f * dpos;
        for (int j = 0; j < i; ++j) {
            float lij = tanhf(lrow[base + j]) * off_scale;
            acc -= (double)lij * y[j];
        }
        double dsafe = fmax((double)dpos, 1e-6);
        y[i] = acc / dsafe;
        logdet += log(dsafe);
        double yc = fmin(fmax(y[i], -1e6), 1e6);
        quad += yc * yc;
        base += i + 1;
    }
    double out64 = -0.5 * (quad + (double)DD * LOG2PI) - logdet;
    comp[idx] = (float)out64;
}

// ---------------------------------------------------------------------
// Per-b: log_softmax(logits) + comp, then logsumexp over K (fp32).
// ---------------------------------------------------------------------
__global__ void combine_kernel(const float* __restrict__ logits,
                               const float* __restrict__ comp,
                               float* __restrict__ out) {
    int b = blockIdx.x * blockDim.x + threadIdx.x;
    if (b >= BB) return;
    float lg[KK];
    float m = -INFINITY;
    for (int k = 0; k < KK; ++k) { lg[k] = logits[b * KK + k]; m = fmaxf(m, lg[k]); }
    float se = 0.0f;
    for (int k = 0; k < KK; ++k) se += __expf(lg[k] - m);
    float lse = logf(se);
    float t[KK];
    float m2 = -INFINITY;
    for (int k = 0; k < KK; ++k) {
        t[k] = (lg[k] - m - lse) + comp[b * KK + k];
        m2 = fmaxf(m2, t[k]);
    }
    float se2 = 0.0f;
    for (int k = 0; k < KK; ++k) se2 += __expf(t[k] - m2);
    out[b] = m2 + logf(se2);
}

// ---------------------------------------------------------------------
// Launch
// ---------------------------------------------------------------------
extern "C" void kernel_launch(void* const* d_in, const int* in_sizes, int n_in,
                              void* d_out, int out_size, void* d_ws, size_t ws_size,
                              hipStream_t stream) {
    const float* s   = (const float*)d_in[0];
    const float* a   = (const float*)d_in[1];
    const float* sp  = (const float*)d_in[2];
    const float* W1  = (const float*)d_in[3];
    const float* b1  = (const float*)d_in[4];
    const float* W2  = (const float*)d_in[5];
    const float* b2  = (const float*)d_in[6];
    const float* W3  = (const float*)d_in[7];
    const float* b3  = (const float*)d_in[8];
    const float* Wmu = (const float*)d_in[9];
    const float* bmu = (const float*)d_in[10];
    const float* WL  = (const float*)d_in[11];
    const float* bL  = (const float*)d_in[12];
    const float* Wpi = (const float*)d_in[13];
    const float* bpi = (const float*)d_in[14];
    float* out = (float*)d_out;

    // workspace carve-up
    char* ws = (char*)d_ws;
    float* X      = (float*)ws;   ws += (size_t)BB * INDIM  * 4;
    float* h1     = (float*)ws;   ws += (size_t)BB * HID    * 4;
    float* h2     = (float*)ws;   ws += (size_t)BB * HID    * 4;
    float* h3     = (float*)ws;   ws += (size_t)BB * HID    * 4;
    float* mu     = (float*)ws;   ws += (size_t)BB * OUT_MU * 4;
    float* Lv     = (float*)ws;   ws += (size_t)BB * OUT_L  * 4;
    float* logits = (float*)ws;   ws += (size_t)BB * KK     * 4;
    float* comp   = (float*)ws;   ws += (size_t)BB * KK     * 4;

    // 1) pack X = [s, a]
    {
        int n = BB * INDIM;
        pack_x_kernel<<<(n + 255) / 256, 256, 0, stream>>>(s, a, X);
    }

    // 2-6) WMMA GEMMs: grid.x = ceil(Ntiles/4), grid.y = Mtiles/4
    const int MB = (BB / 16) / 4;   // 128 M blocks
    auto nbx = [](int N) { return (N / 16 + 3) / 4; };
    gemm_wmma_f32_kernel<<<dim3(nbx(HID),    MB), 128, 0, stream>>>(X,  W1,  b1,  h1, HID,    INDIM, 1);
    gemm_wmma_f32_kernel<<<dim3(nbx(HID),    MB), 128, 0, stream>>>(h1, W2,  b2,  h2, HID,    HID,   1);
    gemm_wmma_f32_kernel<<<dim3(nbx(HID),    MB), 128, 0, stream>>>(h2, W3,  b3,  h3, HID,    HID,   1);
    gemm_wmma_f32_kernel<<<dim3(nbx(OUT_MU), MB), 128, 0, stream>>>(h3, Wmu, bmu, mu, OUT_MU, HID,   0);
    gemm_wmma_f32_kernel<<<dim3(nbx(OUT_L),  MB), 128, 0, stream>>>(h3, WL,  bL,  Lv, OUT_L,  HID,   0);

    // 7) logits head (K=5, VALU)
    {
        int n = BB * KK;
        logits_kernel<<<(n + 255) / 256, 256, 0, stream>>>(h3, Wpi, bpi, logits);
    }

    // 8) per-(b,k) fused L-construction + fp64 triangular solve
    {
        int n = BB * KK;
        mdn_comp_kernel<<<(n + 255) / 256, 256, 0, stream>>>(sp, mu, Lv, comp);
    }

    // 9) log_softmax + logsumexp
    combine_kernel<<<(BB + 255) / 256, 256, 0, stream>>>(logits, comp, out);
}